// MultiheadAttention_15771119911397
// MI455X (gfx1250) — compile-verified
//
#include <hip/hip_runtime.h>
#include <hip/hip_bf16.h>

#define DIM   4096
#define NHEAD 32
#define HD    128
#define HALF  64
#define SEQ   8191      // cached positions
#define S_ALL 8192      // cached + current token
#define NSPLIT 32
#define CHUNK  256      // S_ALL / NSPLIT
#define EPS 1e-6f

typedef float v2f __attribute__((ext_vector_type(2)));
typedef float v8f __attribute__((ext_vector_type(8)));

// workspace layout (float offsets)
#define WS_XN  0
#define WS_Q   (WS_XN + DIM)
#define WS_K   (WS_Q + DIM)
#define WS_V   (WS_K + DIM)
#define WS_M   (WS_V + DIM)                 // NHEAD*NSPLIT
#define WS_L   (WS_M + NHEAD*NSPLIT)        // NHEAD*NSPLIT
#define WS_ACC (WS_L + NHEAD*NSPLIT)        // NHEAD*NSPLIT*HD

// ---------------- RMSNorm: xn = w * x / sqrt(mean(x^2)+eps) ----------------
__global__ void k_rmsnorm(const float* __restrict__ x, const float* __restrict__ w,
                          float* __restrict__ xn) {
  __shared__ float red[256];
  int t = threadIdx.x;
  float s = 0.f;
  for (int i = t; i < DIM; i += 256) { float v = x[i]; s += v * v; }
  red[t] = s; __syncthreads();
  for (int off = 128; off > 0; off >>= 1) {
    if (t < off) red[t] += red[t + off];
    __syncthreads();
  }
  float rinv = rsqrtf(red[0] / (float)DIM + EPS);
  for (int i = t; i < DIM; i += 256) xn[i] = w[i] * x[i] * rinv;
}

// ---------------- q/k/v GEMV via V_WMMA_F32_16X16X4_F32 ----------------
// One wave per 16 output rows. A = 16 weight rows x 4 k-elems; B = xn
// broadcast across all 16 columns, so every column of D is the dot product.
__global__ void k_qkv_gemv(const float* __restrict__ xn,
                           const float* __restrict__ wq,
                           const float* __restrict__ wk,
                           const float* __restrict__ wv,
                           float* __restrict__ ws) {
  int wid  = blockIdx.x * (blockDim.x >> 5) + (threadIdx.x >> 5);
  int lane = threadIdx.x & 31;
  int g    = wid << 4;                 // global output row base (0..12287)
  int mat  = g >> 12;                  // 0=q 1=k 2=v
  int r    = g & (DIM - 1);
  const float* W = (mat == 0) ? wq : (mat == 1) ? wk : wv;
  float* out = ws + ((mat == 0) ? WS_Q : (mat == 1) ? WS_K : WS_V) + r;

  int m   = lane & 15;                 // matrix row within tile
  int h16 = lane >> 4;                 // 0: k,k+1   1: k+2,k+3
  const float* arow = W + (size_t)(r + m) * DIM + 2 * h16;
  const float* bvec = xn + 2 * h16;

  v8f acc = {};
  for (int k = 0; k < DIM; k += 16) {
    v2f a0 = *(const v2f*)(arow + k);
    v2f a1 = *(const v2f*)(arow + k + 4);
    v2f a2 = *(const v2f*)(arow + k + 8);
    v2f a3 = *(const v2f*)(arow + k + 12);
    v2f b0 = *(const v2f*)(bvec + k);
    v2f b1 = *(const v2f*)(bvec + k + 4);
    v2f b2 = *(const v2f*)(bvec + k + 8);
    v2f b3 = *(const v2f*)(bvec + k + 12);
    acc = __builtin_amdgcn_wmma_f32_16x16x4_f32(false, a0, false, b0, (short)0, acc, false, false);
    acc = __builtin_amdgcn_wmma_f32_16x16x4_f32(false, a1, false, b1, (short)0, acc, false, false);
    acc = __builtin_amdgcn_wmma_f32_16x16x4_f32(false, a2, false, b2, (short)0, acc, false, false);
    acc = __builtin_amdgcn_wmma_f32_16x16x4_f32(false, a3, false, b3, (short)0, acc, false, false);
  }
  // D layout: lane 0 col holds M=j in VGPR j; lane 16 holds M=8+j
  if (lane == 0)  { for (int j = 0; j < 8; ++j) out[j]     = acc[j]; }
  if (lane == 16) { for (int j = 0; j < 8; ++j) out[8 + j] = acc[j]; }
}

// ---------------- RoPE on q and k (in place in workspace) ----------------
__global__ void k_rope(float* __restrict__ ws, const float* __restrict__ sinc,
                       const float* __restrict__ cosc, const int* __restrict__ posp) {
  int t = blockIdx.x * blockDim.x + threadIdx.x;   // 0..4095
  int which = t >> 11;                              // 0 = q, 1 = k
  int idx = t & 2047;
  int h = idx >> 6;
  int i = idx & 63;
  int pos = posp[0];
  float s = sinc[pos * HD + i];
  float c = cosc[pos * HD + i];
  float* buf = ws + ((which == 0) ? WS_Q : WS_K) + h * HD;
  float a = buf[i], b = buf[i + HALF];
  buf[i]        = a * c - b * s;
  buf[i + HALF] = a * s + b * c;
}

// ---------------- attention partials (flash-decoding split) ----------------
// Block = (head h, split c of 256 positions). Scores via WMMA tiles of 16
// positions; position SEQ (the new token) reads k/v from workspace.
__global__ void k_attn_part(const float* __restrict__ kc, const float* __restrict__ vc,
                            float* __restrict__ ws) {
  const int h = blockIdx.x;
  const int c = blockIdx.y;
  const int t = threadIdx.x;          // 0..255
  __shared__ float sq[HD];
  __shared__ float ssc[CHUNK];
  __shared__ float st[CHUNK];
  __shared__ float sacc[HD];

  if (t < HD) sq[t] = ws[WS_Q + h * HD + t];
  __syncthreads();

  int wv8  = t >> 5;                   // wave id 0..7
  int lane = t & 31;
  int m    = lane & 15;
  int h16  = lane >> 4;
  for (int tile = wv8 * 2; tile < wv8 * 2 + 2; ++tile) {
    int srow = c * CHUNK + tile * 16 + m;
    const float* arow = (srow < SEQ) ? (kc + (size_t)srow * DIM + h * HD)
                                     : (ws + WS_K + h * HD);
    arow += 2 * h16;
    const float* bvec = sq + 2 * h16;
    v8f acc = {};
    for (int d = 0; d < HD; d += 16) {
      v2f a0 = *(const v2f*)(arow + d);
      v2f a1 = *(const v2f*)(arow + d + 4);
      v2f a2 = *(const v2f*)(arow + d + 8);
      v2f a3 = *(const v2f*)(arow + d + 12);
      v2f b0 = *(const v2f*)(bvec + d);
      v2f b1 = *(const v2f*)(bvec + d + 4);
      v2f b2 = *(const v2f*)(bvec + d + 8);
      v2f b3 = *(const v2f*)(bvec + d + 12);
      acc = __builtin_amdgcn_wmma_f32_16x16x4_f32(false, a0, false, b0, (short)0, acc, false, false);
      acc = __builtin_amdgcn_wmma_f32_16x16x4_f32(false, a1, false, b1, (short)0, acc, false, false);
      acc = __builtin_amdgcn_wmma_f32_16x16x4_f32(false, a2, false, b2, (short)0, acc, false, false);
      acc = __builtin_amdgcn_wmma_f32_16x16x4_f32(false, a3, false, b3, (short)0, acc, false, false);
    }
    int base = tile * 16;
    if (lane == 0)  { for (int j = 0; j < 8; ++j) ssc[base + j]     = acc[j]; }
    if (lane == 16) { for (int j = 0; j < 8; ++j) ssc[base + 8 + j] = acc[j]; }
  }
  __syncthreads();

  // local max
  st[t] = ssc[t]; __syncthreads();
  for (int off = 128; off > 0; off >>= 1) {
    if (t < off) st[t] = fmaxf(st[t], st[t + off]);
    __syncthreads();
  }
  float mloc = st[0];
  __syncthreads();

  // exp + local sum
  float e = __expf(ssc[t] - mloc);
  ssc[t] = e;
  st[t] = e; __syncthreads();
  for (int off = 128; off > 0; off >>= 1) {
    if (t < off) st[t] += st[t + off];
    __syncthreads();
  }
  float lloc = st[0];

  // partial V accumulation: threads split the 256 positions in two halves
  int half = t >> 7, d = t & 127;
  float acc = 0.f;
  for (int sl = half * 128; sl < half * 128 + 128; ++sl) {
    int s = c * CHUNK + sl;
    const float* vrow = (s < SEQ) ? (vc + (size_t)s * DIM + h * HD)
                                  : (ws + WS_V + h * HD);
    acc += ssc[sl] * vrow[d];
  }
  if (half) sacc[d] = acc;
  __syncthreads();
  if (t < HD) {
    ws[WS_ACC + (size_t)(h * NSPLIT + c) * HD + t] = acc + sacc[t];
    if (t == 0) {
      ws[WS_M + h * NSPLIT + c] = mloc;
      ws[WS_L + h * NSPLIT + c] = lloc;
    }
  }
}

// ---------------- merge splits (log-sum-exp) ----------------
__global__ void k_attn_reduce(float* __restrict__ out, const float* __restrict__ ws) {
  int h = blockIdx.x;
  int d = threadIdx.x;                 // 0..127
  const float* mval = ws + WS_M + h * NSPLIT;
  const float* lval = ws + WS_L + h * NSPLIT;
  float M = -3.402823466e38f;
  for (int i = 0; i < NSPLIT; ++i) M = fmaxf(M, mval[i]);
  float L = 0.f, acc = 0.f;
  for (int i = 0; i < NSPLIT; ++i) {
    float sc = __expf(mval[i] - M);
    L += lval[i] * sc;
    acc += ws[WS_ACC + (size_t)(h * NSPLIT + i) * HD + d] * sc;
  }
  out[h * HD + d] = acc / L;
}

extern "C" void kernel_launch(void* const* d_in, const int* in_sizes, int n_in,
                              void* d_out, int out_size, void* d_ws, size_t ws_size,
                              hipStream_t stream) {
  (void)in_sizes; (void)n_in; (void)out_size; (void)ws_size;
  const float* x     = (const float*)d_in[0];
  const int*   pos   = (const int*)d_in[1];
  const float* w_rms = (const float*)d_in[2];
  const float* wq    = (const float*)d_in[3];
  const float* wk    = (const float*)d_in[4];
  const float* wv    = (const float*)d_in[5];
  const float* kc    = (const float*)d_in[6];
  const float* vc    = (const float*)d_in[7];
  const float* sinc  = (const float*)d_in[8];
  const float* cosc  = (const float*)d_in[9];
  float* out = (float*)d_out;
  float* ws  = (float*)d_ws;

  k_rmsnorm<<<1, 256, 0, stream>>>(x, w_rms, ws + WS_XN);
  k_qkv_gemv<<<96, 256, 0, stream>>>(ws + WS_XN, wq, wk, wv, ws);   // 768 waves x 16 rows
  k_rope<<<16, 256, 0, stream>>>(ws, sinc, cosc, pos);
  dim3 gd(NHEAD, NSPLIT);
  k_attn_part<<<gd, 256, 0, stream>>>(kc, vc, ws);
  k_attn_reduce<<<NHEAD, HD, 0, stream>>>(out, ws);
}